// StyleMultiGranularityHeteroGraph_66133906424422
// MI455X (gfx1250) — compile-verified
//
#include <hip/hip_runtime.h>
#include <hip/hip_bf16.h>

typedef __attribute__((ext_vector_type(2))) float v2f;
typedef __attribute__((ext_vector_type(8))) float v8f;

#define NCV 2000
#define NSN 50000
#define NWD 100000
#define CH  256
#define KBLK 64

// ---------------------------------------------------------------------------
// f32 WMMA GEMM:  out[M,256] (+)= rowscale(A[M,K]) @ W[256,K]^T (+bias)
// Block = 512 threads = 16 waves; one block computes a full 16x256 output
// stripe so each A row is read from HBM exactly once (A >> L2 for the big
// projections). A is staged 16xKBLK into LDS per chunk (ds_load_b64 frags,
// 16x reuse); W frags come from global (<=1.25MB, L2/L1 resident).
// M multiples of 16, K multiples of KBLK guaranteed (1280/768/256).
// ---------------------------------------------------------------------------
__global__ void __launch_bounds__(512)
gemm_wmma_f32(const float* __restrict__ A,
              const float* __restrict__ W,
              const float* __restrict__ bias,
              const float* __restrict__ cnt,
              float* __restrict__ out,
              int M, int K, int accumulate)
{
    __shared__ float As[16 * KBLK];

    const int lane = threadIdx.x & 31;
    const int wave = threadIdx.x >> 5;          // 0..15 -> column tile
    const int row0 = blockIdx.x * 16;
    const int col0 = wave * 16;

    const int half = lane >> 4;                 // 0: K+0..1 | 1: K+2..3
    const int l15  = lane & 15;

    // cooperative A-stage mapping: thread t loads 2 floats of the 16xKBLK tile
    const int sr = threadIdx.x >> 5;            // row 0..15
    const int sc = (threadIdx.x & 31) * 2;      // col 0..62
    const float* agrow = A + (size_t)(row0 + sr) * K + sc;

    // W frag: lane l15 holds output col (col0+l15) == row of W
    const float* wrow = W + (size_t)(col0 + l15) * K + half * 2;

    v8f acc = {};
    for (int kc = 0; kc < K; kc += KBLK) {
        *(v2f*)&As[sr * KBLK + sc] = *(const v2f*)(agrow + kc);
        __syncthreads();
#pragma unroll
        for (int k = 0; k < KBLK; k += 4) {
            v2f a = *(const v2f*)&As[l15 * KBLK + k + half * 2];
            v2f b = *(const v2f*)(wrow + kc + k);
            acc = __builtin_amdgcn_wmma_f32_16x16x4_f32(
                /*neg_a=*/false, a, /*neg_b=*/false, b,
                /*c_mod=*/(short)0, acc, /*reuse_a=*/false, /*reuse_b=*/false);
        }
        __syncthreads();
    }

    const float bval  = bias ? bias[col0 + l15] : 0.0f;
    const int   mbase = row0 + half * 8;        // C/D layout: vgpr v -> row v (+8 hi)
#pragma unroll
    for (int v = 0; v < 8; ++v) {
        const int m = mbase + v;
        float s = 1.0f;
        if (cnt) s = 1.0f / fmaxf(cnt[m], 1.0f);
        const float  val = acc[v] * s + bval;
        const size_t idx = (size_t)m * CH + col0 + l15;
        if (accumulate) out[idx] += val;
        else            out[idx]  = val;
    }
}

// ---------------------------------------------------------------------------
// Edge scatter: one wave per edge; 256-wide feature add via f32 atomics.
// h features stay L2-resident (hs+hw = 154MB < 192MB L2) so the gather and
// the atomic adds resolve in L2, not HBM.
// ---------------------------------------------------------------------------
__global__ void scatter_sum(const float* __restrict__ h,
                            const int* __restrict__ ei, int E,
                            float* __restrict__ agg, float* __restrict__ cnt)
{
    const int edge = blockIdx.x * (blockDim.x >> 5) + (threadIdx.x >> 5);
    const int lane = threadIdx.x & 31;
    if (edge >= E) return;
    const int src = ei[edge];
    const int dst = ei[E + edge];
    const float* s = h   + (size_t)src * CH;
    float*       a = agg + (size_t)dst * CH;
#pragma unroll
    for (int i = 0; i < 8; ++i)
        atomicAdd(&a[lane + i * 32], s[lane + i * 32]);
    if (lane == 0) atomicAdd(&cnt[dst], 1.0f);
}

__global__ void zero_f32(float* __restrict__ p, size_t n)
{
    size_t i = (size_t)blockIdx.x * blockDim.x + threadIdx.x;
    if (i < n) p[i] = 0.0f;
}

// Pre-sum self-loop weights/biases per destination node type:
//   sentence: Wr[0]+Wr[1]+Wr[5], bl[0]+bl[1]+bl[5]
//   word:     Wr[2]+Wr[3],       bl[2]+bl[3]
__global__ void sum_self_weights(const float* __restrict__ Wr,
                                 const float* __restrict__ bl,
                                 float* __restrict__ WrS, float* __restrict__ WrW,
                                 float* __restrict__ blS, float* __restrict__ blW)
{
    const int i = blockIdx.x * blockDim.x + threadIdx.x;
    if (i < CH * CH) {
        WrS[i] = Wr[0 * CH * CH + i] + Wr[1 * CH * CH + i] + Wr[5 * CH * CH + i];
        WrW[i] = Wr[2 * CH * CH + i] + Wr[3 * CH * CH + i];
    }
    if (i < CH) {
        blS[i] = bl[0 * CH + i] + bl[1 * CH + i] + bl[5 * CH + i];
        blW[i] = bl[2 * CH + i] + bl[3 * CH + i];
    }
}

extern "C" void kernel_launch(void* const* d_in, const int* in_sizes, int n_in,
                              void* d_out, int out_size, void* d_ws, size_t ws_size,
                              hipStream_t stream)
{
    const float* x_conv = (const float*)d_in[0];
    const float* x_sent = (const float*)d_in[1];
    const float* x_word = (const float*)d_in[2];
    const int*   ei_cs  = (const int*)d_in[3];
    const int*   ei_ss  = (const int*)d_in[4];
    const int*   ei_sw  = (const int*)d_in[5];
    const int*   ei_ww  = (const int*)d_in[6];
    const int*   ei_sc  = (const int*)d_in[7];
    const int*   ei_ws  = (const int*)d_in[8];
    const float* W_conv = (const float*)d_in[9];
    const float* b_conv = (const float*)d_in[10];
    const float* W_sent = (const float*)d_in[11];
    const float* b_sent = (const float*)d_in[12];
    const float* W_word = (const float*)d_in[13];
    const float* b_word = (const float*)d_in[14];
    const float* Wl     = (const float*)d_in[15];
    const float* bl     = (const float*)d_in[16];
    const float* Wr     = (const float*)d_in[17];

    const int E_cs = in_sizes[3] / 2, E_ss = in_sizes[4] / 2, E_sw = in_sizes[5] / 2;
    const int E_ww = in_sizes[6] / 2, E_sc = in_sizes[7] / 2, E_ws = in_sizes[8] / 2;

    // ---- workspace carve (floats) ----
    float* ws  = (float*)d_ws;
    float* hc  = ws; ws += (size_t)NCV * CH;
    float* hs  = ws; ws += (size_t)NSN * CH;
    float* hw  = ws; ws += (size_t)NWD * CH;
    float* agg = ws; ws += (size_t)NWD * CH;   // reused across relations
    float* cnt = ws; ws += (size_t)NWD;
    float* WrS = ws; ws += (size_t)CH * CH;
    float* WrW = ws; ws += (size_t)CH * CH;
    float* blS = ws; ws += (size_t)CH;
    float* blW = ws; ws += (size_t)CH;

    float* out_conv = (float*)d_out;
    float* out_sent = out_conv + (size_t)NCV * CH;
    float* out_word = out_sent + (size_t)NSN * CH;

    const dim3 gblk(512, 1, 1);                // 16 waves -> 16 rows x 256 cols
    auto ggrid = [](int M) { return dim3((unsigned)(M / 16), 1, 1); };

    // 0) pre-summed self weights
    sum_self_weights<<<dim3(CH * CH / 256), dim3(256), 0, stream>>>(Wr, bl, WrS, WrW, blS, blW);

    // 1) node projections (f32 WMMA GEMMs)
    gemm_wmma_f32<<<ggrid(NCV), gblk, 0, stream>>>(x_conv, W_conv, b_conv, nullptr, hc, NCV, 1280, 0);
    gemm_wmma_f32<<<ggrid(NSN), gblk, 0, stream>>>(x_sent, W_sent, b_sent, nullptr, hs, NSN, 1280, 0);
    gemm_wmma_f32<<<ggrid(NWD), gblk, 0, stream>>>(x_word, W_word, b_word, nullptr, hw, NWD, 768, 0);

    // 2) self terms initialize outputs: out = h_dst @ (ΣWr)^T + Σbl
    gemm_wmma_f32<<<ggrid(NCV), gblk, 0, stream>>>(hc, Wr + 4 * CH * CH, bl + 4 * CH, nullptr, out_conv, NCV, CH, 0);
    gemm_wmma_f32<<<ggrid(NSN), gblk, 0, stream>>>(hs, WrS, blS, nullptr, out_sent, NSN, CH, 0);
    gemm_wmma_f32<<<ggrid(NWD), gblk, 0, stream>>>(hw, WrW, blW, nullptr, out_word, NWD, CH, 0);

    // 3) per-relation: zero agg/cnt -> scatter -> mean-GEMM accumulate into out
    auto run_rel = [&](const float* hsrc, const int* ei, int E, int n_dst,
                       int rel, float* outdst) {
        const size_t an = (size_t)n_dst * CH;
        zero_f32<<<dim3((unsigned)((an + 255) / 256)), dim3(256), 0, stream>>>(agg, an);
        zero_f32<<<dim3((unsigned)((n_dst + 255) / 256)), dim3(256), 0, stream>>>(cnt, (size_t)n_dst);
        scatter_sum<<<dim3((unsigned)((E + 7) / 8)), dim3(256), 0, stream>>>(hsrc, ei, E, agg, cnt);
        gemm_wmma_f32<<<ggrid(n_dst), gblk, 0, stream>>>(agg, Wl + (size_t)rel * CH * CH,
                                                         nullptr, cnt, outdst, n_dst, CH, 1);
    };

    run_rel(hc, ei_cs, E_cs, NSN, 0, out_sent);   // conv -> sentence
    run_rel(hs, ei_ss, E_ss, NSN, 1, out_sent);   // sentence -> sentence
    run_rel(hw, ei_ws, E_ws, NSN, 5, out_sent);   // word -> sentence
    run_rel(hs, ei_sw, E_sw, NWD, 2, out_word);   // sentence -> word
    run_rel(hw, ei_ww, E_ww, NWD, 3, out_word);   // word -> word
    run_rel(hs, ei_sc, E_sc, NCV, 4, out_conv);   // sentence -> conv
}